// GraphSAGE_8177617732123
// MI455X (gfx1250) — compile-verified
//
#include <hip/hip_runtime.h>

typedef float v2f __attribute__((ext_vector_type(2)));
typedef float v8f __attribute__((ext_vector_type(8)));
typedef unsigned int v4u __attribute__((ext_vector_type(4)));
typedef int v4i __attribute__((ext_vector_type(4)));
typedef int v8i __attribute__((ext_vector_type(8)));

#define DIM 128          // feature dim D
#define K2 256           // concat dim 2D
#define NSAMP 25         // neighbor samples
#define ROWS_PER_BLOCK 128
#define BN_EPS 1e-5f
#define NORM_EPS 1e-6f

// ---------------------------------------------------------------------------
// Kernel A: fused gather/mean-aggregate + WMMA GEMM (h @ W.T + b) + ReLU.
// One 256-thread block (8 wave32) handles 128 output rows.
// LDS: W staged by TDM (128x256 f32 = 128KB) + h tile (128x256 f32 = 128KB).
// ---------------------------------------------------------------------------
__global__ void __launch_bounds__(256) sage_fused_kernel(
    const float* __restrict__ feats,
    const int*   __restrict__ self_idx,
    const int*   __restrict__ neigh_idx,
    const float* __restrict__ W,      // [128, 256] row-major (j, k)
    const float* __restrict__ bias,   // [128]
    float*       __restrict__ out,    // [N, 128] pre-BN relu output
    int N)
{
    extern __shared__ float smem[];
    float* w_lds = smem;              // [128][256]
    float* h_lds = smem + DIM * K2;   // [128][256]

    const int tid  = threadIdx.x;
    const int lane = tid & 31;
    const int wid  = tid >> 5;
    const int rowBase = blockIdx.x * ROWS_PER_BLOCK + wid * 16;

    // ---- Phase 0: async-DMA W into LDS via the Tensor Data Mover.
    // Each wave issues one TDM descriptor for a 16x256 f32 slab of W.
    {
        const unsigned long long ga =
            (unsigned long long)(const void*)(W + (size_t)wid * 16 * K2);
        const unsigned int laddr =
            (unsigned int)(unsigned long long)(void*)(w_lds + (size_t)wid * 16 * K2);
        v4u g0;
        g0[0] = 1u;                                  // count=1, user descriptor
        g0[1] = laddr;                               // lds_addr (bytes)
        g0[2] = (unsigned int)ga;                    // global_addr[31:0]
        g0[3] = ((unsigned int)(ga >> 32) & 0x01FFFFFFu) | (2u << 30); // [56:32] | type=2
        v8i g1;
        g1[0] = (2 << 16);                           // wg_mask=0, data_size=4B
        g1[1] = (int)((K2 & 0xFFFF) << 16);          // tensor_dim0=256 (lo16)
        g1[2] = (int)((K2 >> 16) | (16 << 16));      // dim0 hi | tensor_dim1=16 (lo16)
        g1[3] = (int)((K2 & 0xFFFF) << 16);          // dim1 hi=0 | tile_dim0=256
        g1[4] = 16;                                  // tile_dim1=16, tile_dim2=0
        g1[5] = K2;                                  // tensor_dim0_stride=256
        g1[6] = 0;
        g1[7] = 0;
        v4i g2 = {0, 0, 0, 0};                       // 2D tensor: groups 2/3 unused
        v4i g3 = {0, 0, 0, 0};
#if defined(__clang_major__) && __clang_major__ >= 23
        v8i g4 = {0, 0, 0, 0, 0, 0, 0, 0};
        __builtin_amdgcn_tensor_load_to_lds(g0, g1, g2, g3, g4, 0);
#else
        __builtin_amdgcn_tensor_load_to_lds(g0, g1, g2, g3, 0);
#endif
    }

    // ---- Phase 1 (overlapped with the TDM): gather self features and
    // mean-aggregate 25 neighbor rows into the h tile.
    // Each wave builds its own 16 rows; 32 lanes x 4 floats cover D=128.
    for (int r = 0; r < 16; ++r) {
        const int row = rowBase + r;
        float sx = 0.f, sy = 0.f, sz = 0.f, sw = 0.f;
        float ax = 0.f, ay = 0.f, az = 0.f, aw = 0.f;
        if (row < N) {
            const size_t si = (size_t)self_idx[row];
            const float4 sv = *(const float4*)(feats + si * DIM + lane * 4);
            sx = sv.x; sy = sv.y; sz = sv.z; sw = sv.w;
            const int* nrow = neigh_idx + (size_t)row * NSAMP;
            for (int s = 0; s < NSAMP; ++s) {
                const size_t ni = (size_t)nrow[s];
                const float4 fv = *(const float4*)(feats + ni * DIM + lane * 4);
                ax += fv.x; ay += fv.y; az += fv.z; aw += fv.w;
            }
            const float inv = 1.0f / (float)NSAMP;
            ax *= inv; ay *= inv; az *= inv; aw *= inv;
        }
        float* hrow = h_lds + (size_t)(wid * 16 + r) * K2;
        *(float4*)(hrow + lane * 4)       = make_float4(sx, sy, sz, sw); // self
        *(float4*)(hrow + DIM + lane * 4) = make_float4(ax, ay, az, aw); // agg
    }

    __builtin_amdgcn_s_wait_tensorcnt(0);  // W slab landed in LDS
    __syncthreads();                       // all waves: W + full h tile visible

    // ---- Phase 2: GEMM via V_WMMA_F32_16X16X4_F32.
    // out[m, j] = sum_k h[m,k] * W[j,k]  (B tile = W rows, already k-major).
    // A frag (16x4 f32): lane m = lane&15, k-pair = (lane>>4)*2 -> 2 f32/lane.
    // B frag (4x16):     lane n = lane&15, same k-pair          -> 2 f32/lane.
    const int mn   = lane & 15;
    const int koff = (lane >> 4) << 1;
    const float* ha = h_lds + (size_t)(wid * 16 + mn) * K2;

    v8f acc[8];
#pragma unroll
    for (int jt = 0; jt < 8; ++jt) acc[jt] = (v8f){0.f,0.f,0.f,0.f,0.f,0.f,0.f,0.f};

    for (int kb = 0; kb < K2; kb += 4) {
        const v2f a = *(const v2f*)(ha + kb + koff);   // reused across 8 B tiles
#pragma unroll
        for (int jt = 0; jt < 8; ++jt) {
            const v2f b = *(const v2f*)(w_lds + (size_t)(jt * 16 + mn) * K2 + kb + koff);
            acc[jt] = __builtin_amdgcn_wmma_f32_16x16x4_f32(
                false, a, false, b, (short)0, acc[jt], false, false);
        }
    }

    // ---- Phase 3: bias + ReLU, store pre-BN activations.
    // C/D layout: lane n = lane&15, VGPR r holds M = r + 8*(lane>>4).
#pragma unroll
    for (int jt = 0; jt < 8; ++jt) {
        const int j  = jt * 16 + mn;
        const float bj = bias[j];
#pragma unroll
        for (int r = 0; r < 8; ++r) {
            const int mm  = r + ((lane >> 4) << 3);
            const int row = rowBase + mm;
            float v = acc[jt][r] + bj;
            v = v > 0.f ? v : 0.f;
            if (row < N) out[(size_t)row * DIM + j] = v;
        }
    }
}

// ---------------------------------------------------------------------------
// Kernel B: per-column batch statistics (deterministic tree reduction).
// One block per column j: stats[j] = mean, stats[128+j] = rsqrt(var + eps).
// ---------------------------------------------------------------------------
__global__ void __launch_bounds__(256) col_stats_kernel(
    const float* __restrict__ out, float* __restrict__ stats, int N)
{
    __shared__ float s1[256];
    __shared__ float s2[256];
    const int j = blockIdx.x;
    const int t = threadIdx.x;
    float sum = 0.f, sq = 0.f;
    for (int i = t; i < N; i += 256) {
        const float v = out[(size_t)i * DIM + j];
        sum += v; sq += v * v;
    }
    s1[t] = sum; s2[t] = sq;
    __syncthreads();
    for (int off = 128; off > 0; off >>= 1) {
        if (t < off) { s1[t] += s1[t + off]; s2[t] += s2[t + off]; }
        __syncthreads();
    }
    if (t == 0) {
        const float mu  = s1[0] / (float)N;
        const float var = s2[0] / (float)N - mu * mu;   // biased variance
        stats[j]       = mu;
        stats[DIM + j] = rsqrtf(var + BN_EPS);
    }
}

// ---------------------------------------------------------------------------
// Kernel C: apply batchnorm affine + row-wise L2 normalize, in place.
// One wave32 per row; 4 columns per lane; wave shuffle reduction for ||y||.
// ---------------------------------------------------------------------------
__global__ void __launch_bounds__(256) bn_l2norm_kernel(
    float* __restrict__ out, const float* __restrict__ stats,
    const float* __restrict__ gamma, const float* __restrict__ beta, int N)
{
    const int lane = threadIdx.x & 31;
    const int wid  = threadIdx.x >> 5;
    const int row  = blockIdx.x * 8 + wid;
    if (row >= N) return;

    float* p = out + (size_t)row * DIM;
    const int j0 = lane * 4;
    float y[4];
    float ss = 0.f;
#pragma unroll
    for (int c = 0; c < 4; ++c) {
        const int j = j0 + c;
        const float x  = p[j];
        const float yv = (x - stats[j]) * stats[DIM + j] * gamma[j] + beta[j];
        y[c] = yv;
        ss += yv * yv;
    }
#pragma unroll
    for (int off = 16; off > 0; off >>= 1)
        ss += __shfl_xor(ss, off, 32);
    const float inv = 1.0f / (sqrtf(ss) + NORM_EPS);
#pragma unroll
    for (int c = 0; c < 4; ++c)
        p[j0 + c] = y[c] * inv;
}

// ---------------------------------------------------------------------------
extern "C" void kernel_launch(void* const* d_in, const int* in_sizes, int n_in,
                              void* d_out, int out_size, void* d_ws, size_t ws_size,
                              hipStream_t stream)
{
    const float* feats     = (const float*)d_in[0];  // [200000, 128]
    const int*   self_idx  = (const int*)d_in[1];    // [N]
    const int*   neigh_idx = (const int*)d_in[2];    // [N, 25]
    const float* W         = (const float*)d_in[3];  // [128, 256]
    const float* bias      = (const float*)d_in[4];  // [128]
    const float* gamma     = (const float*)d_in[5];  // [128]
    const float* beta      = (const float*)d_in[6];  // [128]
    float* out   = (float*)d_out;                    // [N, 128]
    float* stats = (float*)d_ws;                     // 256 floats (mu, rstd)

    const int N = in_sizes[1];                       // 50000

    const int blocksA = (N + ROWS_PER_BLOCK - 1) / ROWS_PER_BLOCK;
    const size_t ldsA = 2u * DIM * K2 * sizeof(float); // 256 KB (of 320 KB WGP LDS)
    sage_fused_kernel<<<blocksA, 256, ldsA, stream>>>(
        feats, self_idx, neigh_idx, W, bias, out, N);

    col_stats_kernel<<<DIM, 256, 0, stream>>>(out, stats, N);

    const int blocksC = (N + 7) / 8;
    bn_l2norm_kernel<<<blocksC, 256, 0, stream>>>(out, stats, gamma, beta, N);
}